// RGCNRoot_no_jraph_39290360824135
// MI455X (gfx1250) — compile-verified
//
#include <hip/hip_runtime.h>

#define N_NODES 50000
#define N_EDGES 100000
#define IN_FEATS 63
#define HIDDEN 64
#define NUM_REL 8
#define ROW_STRIDE 66   // 64 cols + pad: distinct LDS banks across 16 rows, keeps 8B align
#define MAX_TILES (N_EDGES / 16 + NUM_REL)   // 6258: per-relation ceil padding bound

typedef __attribute__((ext_vector_type(2))) float v2f;
typedef __attribute__((ext_vector_type(8))) float v8f;

// ---------------- utility kernels ----------------

__global__ void zero_f32(float* __restrict__ p, long n) {
  long i = (long)blockIdx.x * blockDim.x + threadIdx.x;
  long stride = (long)gridDim.x * blockDim.x;
  for (; i < n; i += stride) p[i] = 0.f;
}

// Degrees (float) + relation histogram (int) in one pass over the edge list.
__global__ void degree_hist_kernel(const int* __restrict__ snd, const int* __restrict__ rcv,
                                   const int* __restrict__ et, float* __restrict__ degS,
                                   float* __restrict__ degR, int* __restrict__ relCnt) {
  int e = blockIdx.x * blockDim.x + threadIdx.x;
  if (e < N_EDGES) {
    atomicAdd(&degS[snd[e]], 1.f);
    atomicAdd(&degR[rcv[e]], 1.f);
    atomicAdd(&relCnt[et[e]], 1);
  }
}

// In-place deg -> rsqrt(max(deg,1)); covers deg_s and deg_r (contiguous 2*N_NODES)
__global__ void norm_kernel(float* __restrict__ deg) {
  int i = blockIdx.x * blockDim.x + threadIdx.x;
  if (i < 2 * N_NODES) deg[i] = rsqrtf(fmaxf(deg[i], 1.f));
}

// 8-element exclusive scans: bucket edge offsets and 16-edge tile offsets.
__global__ void offsets_kernel(const int* __restrict__ relCnt, int* __restrict__ bucketOff,
                               int* __restrict__ tileStart) {
  if (threadIdx.x == 0) {
    int bo = 0, ts = 0;
    for (int r = 0; r < NUM_REL; ++r) {
      bucketOff[r] = bo;
      tileStart[r] = ts;
      bo += relCnt[r];
      ts += (relCnt[r] + 15) / 16;
    }
    bucketOff[NUM_REL] = bo;
    tileStart[NUM_REL] = ts;
  }
}

// Counting-sort scatter: perm[] holds edge ids grouped by relation.
__global__ void bucket_kernel(const int* __restrict__ et, const int* __restrict__ bucketOff,
                              int* __restrict__ relCursor, int* __restrict__ perm) {
  int e = blockIdx.x * blockDim.x + threadIdx.x;
  if (e < N_EDGES) {
    int r = et[e];
    int pos = atomicAdd(&relCursor[r], 1);
    perm[bucketOff[r] + pos] = e;
  }
}

// ---------------- WMMA message kernel ----------------
// One wave32 = one tile of 16 same-relation edges (post counting sort).
// A (16x64 f32, col63 zero, rows pre-scaled by norm_s[sender]) staged in LDS.
// C(16x64) = A @ K_r as a straight-line chain of V_WMMA_F32_16X16X4_F32
// (16 K-steps x 4 N-tiles), then f32 atomic scatter into agg[receiver].
// Lane layout (ISA 7.12.2):
//   A 16x4 : lane L -> M = L&15, K = 4*ks + 2*(L>>4), +1
//   B 4x16 : lane L -> N = L&15, K = 4*ks + 2*(L>>4), +1
//   C 16x16: VGPR j, lane L -> M = j + 8*(L>>4), N = L&15
__global__ __launch_bounds__(256) void message_kernel(
    const float* __restrict__ nodes, const int* __restrict__ snd,
    const int* __restrict__ rcv, const float* __restrict__ kern,
    const float* __restrict__ normS, const int* __restrict__ perm,
    const int* __restrict__ bucketOff, const int* __restrict__ tileStart,
    float* __restrict__ agg) {
  __shared__ float AS[8][16][ROW_STRIDE];
  __shared__ int recvS[8][16];
  __shared__ int validS[8][16];
  __shared__ int sendS[8][16];
  __shared__ float nrmS[8][16];

  const int wv = threadIdx.x >> 5;
  const int lane = threadIdx.x & 31;
  const int tile = __builtin_amdgcn_readfirstlane(blockIdx.x * 8 + wv);

  // Warm the 126KB relation-kernel table into cache (global_prefetch_b8).
  for (int i = threadIdx.x * 16; i < NUM_REL * IN_FEATS * HIDDEN; i += 256 * 16)
    __builtin_prefetch(&kern[i], 0, 3);

  const int total = __builtin_amdgcn_readfirstlane(tileStart[NUM_REL]);
  const bool active = tile < total;

  // Map tile -> relation via the 8-entry tile-offset scan (uniform per wave).
  int r = 0;
  for (int i = 1; i < NUM_REL; ++i)
    if (active && tile >= tileStart[i]) r = i;
  r = __builtin_amdgcn_readfirstlane(r);

  const int p0 = __builtin_amdgcn_readfirstlane(
      active ? bucketOff[r] + (tile - tileStart[r]) * 16 : 0);
  const int pend = __builtin_amdgcn_readfirstlane(active ? bucketOff[r + 1] : 0);

  // Stage per-edge metadata (16 permuted, same-relation edges).
  if (lane < 16) {
    int p = p0 + lane;
    bool v = active && (p < pend);
    int e = v ? perm[p] : 0;
    int s = v ? snd[e] : 0;
    sendS[wv][lane] = s;
    recvS[wv][lane] = v ? rcv[e] : 0;
    validS[wv][lane] = v ? 1 : 0;
    nrmS[wv][lane] = v ? normS[s] : 0.f;
  }
  __syncthreads();

  // Stage A tile: 16 rows x 64 cols, row-scaled by norm_s, col 63 / invalid rows zero.
  for (int i = lane; i < 16 * 64; i += 32) {
    int row = i >> 6, col = i & 63;
    float val = 0.f;
    if (col < IN_FEATS && validS[wv][row])
      val = nodes[(long)sendS[wv][row] * (IN_FEATS + 1) + col] * nrmS[wv][row];
    AS[wv][row][col] = val;
  }
  __syncthreads();

  if (!active) return;  // after all barriers; whole wave exits together

  const float* __restrict__ Kr = kern + (long)r * IN_FEATS * HIDDEN;
  const int row = lane & 15;
  const int khalf = lane >> 4;  // 0 or 1
  const int ncol = lane & 15;

  v8f acc[4];
  for (int t = 0; t < 4; ++t)
    for (int j = 0; j < 8; ++j) acc[t][j] = 0.f;

  for (int ks = 0; ks < 16; ++ks) {
    const int k0 = ks * 4 + khalf * 2;
    v2f a;
    a.x = AS[wv][row][k0];
    a.y = AS[wv][row][k0 + 1];
    for (int t = 0; t < 4; ++t) {
      const int n = t * 16 + ncol;
      v2f b;
      b.x = Kr[(long)k0 * HIDDEN + n];  // k0 <= 62: always in range
      b.y = (k0 + 1 < IN_FEATS) ? Kr[(long)(k0 + 1) * HIDDEN + n] : 0.f;
      acc[t] = __builtin_amdgcn_wmma_f32_16x16x4_f32(
          false, a, false, b, (short)0, acc[t], false, false);
    }
  }

  // Scatter: message row m -> agg[receiver[m], :] via f32 atomics.
  for (int t = 0; t < 4; ++t) {
    for (int j = 0; j < 8; ++j) {
      const int m = j + 8 * khalf;
      if (validS[wv][m]) {
        const int n = t * 16 + ncol;
        atomicAdd(&agg[(long)recvS[wv][m] * HIDDEN + n], acc[t][j]);
      }
    }
  }
}

// ---------------- node reduction ----------------
// One wave per node: partial = is_root * sum_f relu(agg[n,f]*norm_r[n]) * w[f]
__global__ void finalize_kernel(const float* __restrict__ agg, const float* __restrict__ normR,
                                const float* __restrict__ nodes, const float* __restrict__ w,
                                float* __restrict__ acc) {
  int wv = threadIdx.x >> 5, lane = threadIdx.x & 31;
  int node = blockIdx.x * 8 + wv;
  if (node >= N_NODES) return;
  float nr = normR[node];
  const float2* a2 = (const float2*)(agg + (long)node * HIDDEN);
  float2 a = a2[lane];
  float p = fmaxf(a.x * nr, 0.f) * w[2 * lane] + fmaxf(a.y * nr, 0.f) * w[2 * lane + 1];
  for (int off = 16; off > 0; off >>= 1) p += __shfl_down(p, off, 32);
  if (lane == 0) {
    float root = nodes[(long)node * (IN_FEATS + 1) + IN_FEATS];
    atomicAdd(acc, p * root);
  }
}

__global__ void writeout_kernel(const float* __restrict__ acc, const float* __restrict__ b,
                                float* __restrict__ out) {
  if (threadIdx.x == 0) out[0] = acc[0] + b[0];
}

// ---------------- launch ----------------

extern "C" void kernel_launch(void* const* d_in, const int* in_sizes, int n_in,
                              void* d_out, int out_size, void* d_ws, size_t ws_size,
                              hipStream_t stream) {
  const float* nodes    = (const float*)d_in[0];
  const int* senders    = (const int*)d_in[1];
  const int* receivers  = (const int*)d_in[2];
  const int* edge_types = (const int*)d_in[3];
  // d_in[4] = n_node (single graph, unused)
  const float* kernels  = (const float*)d_in[5];
  const float* dense_w  = (const float*)d_in[6];
  const float* dense_b  = (const float*)d_in[7];

  // ws layout (4-byte units):
  float* ws   = (float*)d_ws;
  float* degS = ws;                              // [N_NODES]  -> norm_s
  float* degR = ws + N_NODES;                    // [N_NODES]  -> norm_r
  float* agg  = ws + 2 * N_NODES;                // [N_NODES*HIDDEN]
  float* acc  = agg + (long)N_NODES * HIDDEN;    // [1]
  long ibase  = 2L * N_NODES + (long)N_NODES * HIDDEN + 4;  // 16B-aligned int region
  int* relCnt    = (int*)ws + ibase;             // [8]
  int* relCursor = relCnt + NUM_REL;             // [8]
  int* bucketOff = relCursor + NUM_REL;          // [9]
  int* tileStart = bucketOff + NUM_REL + 1;      // [9]
  int* perm      = tileStart + NUM_REL + 1;      // [N_EDGES]

  // Zero floats + relCnt + relCursor (bucketOff/tileStart/perm fully overwritten later).
  long zcount = ibase + 2 * NUM_REL;
  zero_f32<<<2048, 256, 0, stream>>>(ws, zcount);

  degree_hist_kernel<<<(N_EDGES + 255) / 256, 256, 0, stream>>>(
      senders, receivers, edge_types, degS, degR, relCnt);
  norm_kernel<<<(2 * N_NODES + 255) / 256, 256, 0, stream>>>(ws);
  offsets_kernel<<<1, 32, 0, stream>>>(relCnt, bucketOff, tileStart);
  bucket_kernel<<<(N_EDGES + 255) / 256, 256, 0, stream>>>(
      edge_types, bucketOff, relCursor, perm);

  int blocks = (MAX_TILES + 7) / 8;  // 8 waves (tiles) per 256-thread block
  message_kernel<<<blocks, 256, 0, stream>>>(nodes, senders, receivers, kernels,
                                             degS, perm, bucketOff, tileStart, agg);

  finalize_kernel<<<(N_NODES + 7) / 8, 256, 0, stream>>>(agg, degR, nodes, dense_w, acc);
  writeout_kernel<<<1, 32, 0, stream>>>(acc, dense_b, (float*)d_out);
}